// Dct_50869592655315
// MI455X (gfx1250) — compile-verified
//
#include <hip/hip_runtime.h>

// ---------------------------------------------------------------------------
// 2-D DCT-II (ortho) of (B,C,512,512) fp32  ==  Y = D * X * D^T  per image.
// Two batched 512^3 fp32 GEMMs on the CDNA5 matrix core
// (V_WMMA_F32_16X16X4_F32). D and D^T are built once into workspace.
//
// GEMM kernel: block = 128 threads (4 wave32s), block tile 128x64, KC=32,
// double-buffered LDS with register prefetch. Both A (MxK) and B (staged
// transposed, NxK) use stride-36 LDS rows so every WMMA fragment is a single
// conflict-free ds_load_b64 (no VALU marshaling).
// ---------------------------------------------------------------------------

typedef __attribute__((ext_vector_type(2))) float v2f;
typedef __attribute__((ext_vector_type(8))) float v8f;

#define NDIM 512
#define BM   128   // block tile M
#define BN   64    // block tile N
#define KC   32    // K chunk staged in LDS
#define LSTR 36    // LDS row stride (dwords): fragment reads hit banks
                   // =0/1 (mod 4) for lanes 0-15 vs =2/3 for lanes 16-31

// ---------------------------------------------------------------------------
// D[k][n] = s_k * cos(pi*(2n+1)*k / (2N)),  s_0 = 1/sqrt(N), s_k = sqrt(2/N).
// ---------------------------------------------------------------------------
__global__ __launch_bounds__(256) void dct_build(float* __restrict__ D,
                                                 float* __restrict__ DT) {
    int idx = blockIdx.x * blockDim.x + threadIdx.x;
    if (idx >= NDIM * NDIM) return;
    int k = idx >> 9;
    int n = idx & (NDIM - 1);
    const double PI = 3.14159265358979323846;
    double s = (k == 0) ? (1.0 / sqrt((double)NDIM)) : sqrt(2.0 / (double)NDIM);
    double v = s * cos(PI * (double)((2 * n + 1) * k) / (2.0 * (double)NDIM));
    float f = (float)v;
    D[k * NDIM + n]  = f;
    DT[n * NDIM + k] = f;
}

// Fetch one KC-chunk of A (BM x KC, row-major float4) and B (KC x BN, read
// along K so the LDS store of the transposed tile is a contiguous b128).
__device__ __forceinline__ void global_fetch(const float* __restrict__ A,
                                             const float* __restrict__ B,
                                             int m0, int n0, int kb, int tid,
                                             float4 (&ar)[8], float4 (&br)[4]) {
    #pragma unroll
    for (int i = 0; i < 8; ++i) {
        int idx = tid + i * 128;            // 0..1023
        int r   = idx >> 3;                 // row 0..127
        int c   = (idx & 7) << 2;           // col 0,4,..,28
        ar[i] = *reinterpret_cast<const float4*>(&A[(long)(m0 + r) * NDIM + kb + c]);
    }
    const int bn = tid & 63;                // column 0..63
    const int bh = tid >> 6;                // 0/1: low/high K half of each octet
    #pragma unroll
    for (int g = 0; g < 4; ++g) {
        int k4 = bh * 4 + g * 8;            // {0,8,16,24} + 4*bh -> covers 0..28
        float4 v;
        v.x = B[(long)(kb + k4 + 0) * NDIM + n0 + bn];
        v.y = B[(long)(kb + k4 + 1) * NDIM + n0 + bn];
        v.z = B[(long)(kb + k4 + 2) * NDIM + n0 + bn];
        v.w = B[(long)(kb + k4 + 3) * NDIM + n0 + bn];
        br[g] = v;
    }
}

__device__ __forceinline__ void lds_store(float* __restrict__ as,
                                          float* __restrict__ bt, int tid,
                                          const float4 (&ar)[8],
                                          const float4 (&br)[4]) {
    #pragma unroll
    for (int i = 0; i < 8; ++i) {
        int idx = tid + i * 128;
        int r   = idx >> 3;
        int c   = (idx & 7) << 2;
        *reinterpret_cast<float4*>(&as[r * LSTR + c]) = ar[i];
    }
    const int bn = tid & 63;
    const int bh = tid >> 6;
    #pragma unroll
    for (int g = 0; g < 4; ++g) {
        int k4 = bh * 4 + g * 8;
        *reinterpret_cast<float4*>(&bt[bn * LSTR + k4]) = br[g];  // B^T tile
    }
}

// ---------------------------------------------------------------------------
// C[b] = A[b] * B[b]  (512x512 NN, fp32, strided batch; stride 0 = shared).
// grid: (N/BN, M/BM, batch), block: 128 threads = 4 wave32s in a 2x2 grid,
// each wave computing a 64x32 region as 4x2 WMMA 16x16 accumulators.
// ---------------------------------------------------------------------------
__global__ __launch_bounds__(128) void gemm512_nn(const float* __restrict__ A, long sA,
                                                  const float* __restrict__ B, long sB,
                                                  float* __restrict__ C, long sC) {
    __shared__ float As[2][BM * LSTR];   // A tile, row-major (M x K)
    __shared__ float Bt[2][BN * LSTR];   // B tile, transposed  (N x K)

    const int b = blockIdx.z;
    A += (long)b * sA;
    B += (long)b * sB;
    C += (long)b * sC;

    const int m0   = blockIdx.y * BM;
    const int n0   = blockIdx.x * BN;
    const int tid  = threadIdx.x;
    const int lane = tid & 31;
    const int hi   = lane >> 4;          // half-wave: K pair (0,1) vs (2,3)
    const int lq   = lane & 15;          // index within 16-wide fragment
    const int wave = tid >> 5;
    const int wm   = (wave >> 1) * 64;   // wave tile row offset
    const int wn   = (wave & 1) * 32;    // wave tile col offset

    v8f acc[4][2] = {};

    float4 ar[8], br[4];
    global_fetch(A, B, m0, n0, 0, tid, ar, br);
    lds_store(As[0], Bt[0], tid, ar, br);
    __syncthreads();

    for (int kb = 0; kb < NDIM; kb += KC) {
        const int  cur  = (kb >> 5) & 1;
        const bool more = (kb + KC) < NDIM;
        if (more) global_fetch(A, B, m0, n0, kb + KC, tid, ar, br);

        const float* __restrict__ as = As[cur];
        const float* __restrict__ bt = Bt[cur];
        #pragma unroll
        for (int kk = 0; kk < KC; kk += 4) {
            v2f af[4], bf[2];
            #pragma unroll
            for (int i = 0; i < 4; ++i)
                af[i] = *reinterpret_cast<const v2f*>(
                    &as[(wm + i * 16 + lq) * LSTR + kk + 2 * hi]);
            #pragma unroll
            for (int j = 0; j < 2; ++j)
                bf[j] = *reinterpret_cast<const v2f*>(
                    &bt[(wn + j * 16 + lq) * LSTR + kk + 2 * hi]);
            #pragma unroll
            for (int i = 0; i < 4; ++i)
                #pragma unroll
                for (int j = 0; j < 2; ++j)
                    acc[i][j] = __builtin_amdgcn_wmma_f32_16x16x4_f32(
                        false, af[i], false, bf[j],
                        (short)0, acc[i][j], false, false);
        }
        if (more) lds_store(As[cur ^ 1], Bt[cur ^ 1], tid, ar, br);
        __syncthreads();
    }

    // C/D fragment layout: lane<16 -> M = r, lane>=16 -> M = r+8, N = lq.
    #pragma unroll
    for (int i = 0; i < 4; ++i) {
        #pragma unroll
        for (int j = 0; j < 2; ++j) {
            int col = n0 + wn + j * 16 + lq;
            #pragma unroll
            for (int r = 0; r < 8; ++r) {
                int row = m0 + wm + i * 16 + hi * 8 + r;
                C[(long)row * NDIM + col] = acc[i][j][r];
            }
        }
    }
}

extern "C" void kernel_launch(void* const* d_in, const int* in_sizes, int n_in,
                              void* d_out, int out_size, void* d_ws, size_t ws_size,
                              hipStream_t stream) {
    const float* inp = (const float*)d_in[0];
    float* out = (float*)d_out;

    const long img   = (long)NDIM * NDIM;            // 262144
    const int  batch = (int)(in_sizes[0] / img);     // 96 for (32,3,512,512)

    float* D  = (float*)d_ws;        // 512*512 floats
    float* DT = D + img;             // 512*512 floats
    float* T  = DT + img;            // batch * 512*512 floats (intermediate)

    // 1) Build DCT matrix and its transpose.
    dct_build<<<(NDIM * NDIM) / 256, 256, 0, stream>>>(D, DT);

    dim3 grid(NDIM / BN, NDIM / BM, batch);

    // 2) T[b] = X[b] * D^T   (DCT along W)
    gemm512_nn<<<grid, 128, 0, stream>>>(inp, img, DT, 0L, T, img);

    // 3) Y[b] = D * T[b]     (DCT along H)
    gemm512_nn<<<grid, 128, 0, stream>>>(D, 0L, T, img, out, img);
}